// MAdapterBlock_44281112821985
// MI455X (gfx1250) — compile-verified
//
#include <hip/hip_runtime.h>
#include <hip/hip_bf16.h>
#include <math.h>

#define NSEQ   4
#define LSEQ   1024
#define CMODEL 256
#define DINNER 512
#define DCONV  4
#define DTRANK 16
#define DSTATE 16
#define NROWS  (NSEQ * LSEQ)         /* 4096 */
#define DBLW   (DTRANK + 2 * DSTATE) /* 48 */

typedef float v2f __attribute__((ext_vector_type(2)));
typedef float v8f __attribute__((ext_vector_type(8)));

// ---------------------------------------------------------------------------
// (1) x (bn, c, l)  ->  hs (bn*L + l, c)   [coalesced writes]
// ---------------------------------------------------------------------------
__global__ void mad_init_hs(const float* __restrict__ x, float* __restrict__ hs) {
    int idx = blockIdx.x * blockDim.x + threadIdx.x;   // output order
    if (idx >= NROWS * CMODEL) return;
    int c  = idx % CMODEL;
    int l  = (idx / CMODEL) % LSEQ;
    int bn = idx / (CMODEL * LSEQ);
    hs[idx] = x[((size_t)(bn * CMODEL + c)) * LSEQ + l];
}

// ---------------------------------------------------------------------------
// (2) out = alpha*a + beta*b  (in-place safe: pure elementwise)
// ---------------------------------------------------------------------------
__global__ void mad_axpby(float* __restrict__ out, const float* __restrict__ a,
                          const float* __restrict__ b, float alpha, float beta, int n) {
    int i = blockIdx.x * blockDim.x + threadIdx.x;
    if (i < n) out[i] = alpha * a[i] + beta * b[i];
}

// ---------------------------------------------------------------------------
// (3) LayerNorm over C=256 per (bn,l) row.  One 256-thread block per row.
// ---------------------------------------------------------------------------
__global__ void mad_layernorm(const float* __restrict__ resid,
                              const float* __restrict__ w, const float* __restrict__ b,
                              float* __restrict__ hn) {
    __shared__ float ssum[CMODEL];
    __shared__ float ssq[CMODEL];
    int row = blockIdx.x;
    int c   = threadIdx.x;
    float v = resid[(size_t)row * CMODEL + c];
    ssum[c] = v;
    ssq[c]  = v * v;
    __syncthreads();
    for (int s = CMODEL / 2; s > 0; s >>= 1) {
        if (c < s) { ssum[c] += ssum[c + s]; ssq[c] += ssq[c + s]; }
        __syncthreads();
    }
    float mu  = ssum[0] * (1.0f / CMODEL);
    float var = ssq[0] * (1.0f / CMODEL) - mu * mu;
    hn[(size_t)row * CMODEL + c] = (v - mu) * rsqrtf(var + 1e-5f) * w[c] + b[c];
}

// ---------------------------------------------------------------------------
// (4) WMMA fp32 GEMM:  C[M,N] (+)= A[M,K] * W[N,K]^T  (+bias, +softplus)
//     One wave32 computes a 16 x (16*NT) strip; A-frag shared across NT
//     accumulators.  Software-pipelined K-loop: fragments for step k+1 are
//     loaded before the WMMAs of step k so load latency overlaps matrix ops
//     (avoids s_wait_loadcnt 0 immediately before each WMMA).
//     grid = (N/(16*NT), M/16), block = 32 (one wave, EXEC all ones).
//     V_WMMA_F32_16X16X4_F32 fragment layout (ISA 7.12.2):
//       A frag: lane -> m = tileM + lane%16, k = k0 + 2*(lane/16) + {0,1}
//       B frag: lane -> n = tileN + lane%16, k = k0 + 2*(lane/16) + {0,1}
//       D frag: vgpr r -> m = r + 8*(lane/16), n = lane%16
// ---------------------------------------------------------------------------
template <int NT>
__global__ __launch_bounds__(32)
void mad_wmma_gemm(const float* __restrict__ A, int lda,
                   const float* __restrict__ W, int ldw,
                   const float* __restrict__ bias,
                   float* __restrict__ C, int ldc,
                   int K, int act_softplus, int accumulate) {
    int lane = threadIdx.x;       // 0..31
    int hh   = lane >> 4;
    int l16  = lane & 15;
    int tm   = blockIdx.y;
    int tn0  = blockIdx.x * NT;   // first 16-wide N subtile
    int row  = tm * 16 + l16;

    const float* Arow = A + (size_t)row * lda + 2 * hh;
    const float* Wrow[NT];
    v8f acc[NT];
#pragma unroll
    for (int t = 0; t < NT; ++t) {
        int col = (tn0 + t) * 16 + l16;
        Wrow[t] = W + (size_t)col * ldw + 2 * hh;
        if (accumulate) {
#pragma unroll
            for (int r = 0; r < 8; ++r)
                acc[t][r] = C[(size_t)(tm * 16 + r + 8 * hh) * ldc + col];
        } else {
#pragma unroll
            for (int r = 0; r < 8; ++r) acc[t][r] = 0.0f;
        }
    }

    // --- software-pipelined K loop (prologue load, steady state, epilogue) ---
    v2f a_cur = *(const v2f*)(Arow);
    v2f w_cur[NT];
#pragma unroll
    for (int t = 0; t < NT; ++t) w_cur[t] = *(const v2f*)(Wrow[t]);

    for (int k0 = 4; k0 < K; k0 += 4) {
        // issue next-step loads first: they retire while WMMAs below execute
        v2f a_nxt = *(const v2f*)(Arow + k0);
        v2f w_nxt[NT];
#pragma unroll
        for (int t = 0; t < NT; ++t) w_nxt[t] = *(const v2f*)(Wrow[t] + k0);

#pragma unroll
        for (int t = 0; t < NT; ++t)
            acc[t] = __builtin_amdgcn_wmma_f32_16x16x4_f32(
                false, a_cur, false, w_cur[t], (short)0, acc[t], false, false);

        a_cur = a_nxt;
#pragma unroll
        for (int t = 0; t < NT; ++t) w_cur[t] = w_nxt[t];
    }
#pragma unroll
    for (int t = 0; t < NT; ++t)
        acc[t] = __builtin_amdgcn_wmma_f32_16x16x4_f32(
            false, a_cur, false, w_cur[t], (short)0, acc[t], false, false);

    // --- store with optional bias / softplus ---
#pragma unroll
    for (int t = 0; t < NT; ++t) {
        int col = (tn0 + t) * 16 + l16;
        float bv = bias ? bias[col] : 0.0f;
#pragma unroll
        for (int r = 0; r < 8; ++r) {
            float v = acc[t][r] + bv;
            if (act_softplus) v = (v > 20.0f) ? v : log1pf(expf(v));
            C[(size_t)(tm * 16 + r + 8 * hh) * ldc + col] = v;
        }
    }
}

// ---------------------------------------------------------------------------
// (5) depthwise causal conv (logical t-order; dir flips index map) + SiLU
// ---------------------------------------------------------------------------
__global__ void mad_conv_silu(const float* __restrict__ xz,
                              const float* __restrict__ cw, const float* __restrict__ cb,
                              float* __restrict__ xact, int dir) {
    int idx = blockIdx.x * blockDim.x + threadIdx.x;   // over NROWS*DINNER
    if (idx >= NROWS * DINNER) return;
    int ch = idx % DINNER;
    int t  = (idx / DINNER) % LSEQ;
    int bn = idx / (DINNER * LSEQ);
    float acc = cb[ch];
#pragma unroll
    for (int k = 0; k < DCONV; ++k) {
        int tt = t - (DCONV - 1) + k;
        if (tt >= 0) {
            int pp = dir ? (LSEQ - 1 - tt) : tt;
            acc += xz[((size_t)(bn * LSEQ + pp)) * (2 * DINNER) + ch] * cw[ch * DCONV + k];
        }
    }
    int p = dir ? (LSEQ - 1 - t) : t;
    xact[((size_t)(bn * LSEQ + p)) * DINNER + ch] = acc / (1.0f + expf(-acc));
}

// ---------------------------------------------------------------------------
// (6) selective scan: one thread per (bn, ch); 16 states in registers.
//     B/C loads wave-uniform (broadcast); dt/x/z coalesced over ch.
//     y written into the dead x-half of xz (stride 2*DINNER).
// ---------------------------------------------------------------------------
__global__ void mad_scan(const float* __restrict__ xact, const float* __restrict__ dt,
                         const float* __restrict__ dbl, const float* __restrict__ xz,
                         const float* __restrict__ A_log, const float* __restrict__ Dp,
                         float* __restrict__ y, int ystride, int dir) {
    int idx = blockIdx.x * blockDim.x + threadIdx.x;   // over NSEQ*DINNER = 2048
    if (idx >= NSEQ * DINNER) return;
    int ch = idx % DINNER;
    int bn = idx / DINNER;

    float A[DSTATE], h[DSTATE];
#pragma unroll
    for (int n = 0; n < DSTATE; ++n) {
        A[n] = -expf(A_log[ch * DSTATE + n]);
        h[n] = 0.0f;
    }
    float dpv = Dp[ch];

    for (int t = 0; t < LSEQ; ++t) {
        int p = dir ? (LSEQ - 1 - t) : t;
        size_t r = (size_t)bn * LSEQ + p;
        float dtv = dt[r * DINNER + ch];
        float xv  = xact[r * DINNER + ch];
        float zv  = xz[r * (2 * DINNER) + DINNER + ch];
        const float* Brow = dbl + r * DBLW + DTRANK;
        const float* Crow = Brow + DSTATE;
        float yv = 0.0f;
#pragma unroll
        for (int n = 0; n < DSTATE; ++n) {
            h[n] = expf(dtv * A[n]) * h[n] + dtv * Brow[n] * xv;
            yv += h[n] * Crow[n];
        }
        yv = (yv + dpv * xv) * (zv / (1.0f + expf(-zv)));
        y[r * ystride + ch] = yv;
    }
}

// ---------------------------------------------------------------------------
// (7) final: out[bn,c,l] = 2*resid[bn,l,c] + hs[bn,l,c]
// ---------------------------------------------------------------------------
__global__ void mad_final(const float* __restrict__ resid, const float* __restrict__ hs,
                          float* __restrict__ out) {
    int idx = blockIdx.x * blockDim.x + threadIdx.x;   // output order (bn,c,l)
    if (idx >= NROWS * CMODEL) return;
    int l  = idx % LSEQ;
    int c  = (idx / LSEQ) % CMODEL;
    int bn = idx / (LSEQ * CMODEL);
    size_t r = ((size_t)(bn * LSEQ + l)) * CMODEL + c;
    out[idx] = 2.0f * resid[r] + hs[r];
}

// ---------------------------------------------------------------------------
extern "C" void kernel_launch(void* const* d_in, const int* in_sizes, int n_in,
                              void* d_out, int out_size, void* d_ws, size_t ws_size,
                              hipStream_t stream) {
    const float* x        = (const float*)d_in[0];
    const float* norm_w   = (const float*)d_in[1];
    const float* norm_b   = (const float*)d_in[2];
    const float* in_w     = (const float*)d_in[3];
    const float* conv_w   = (const float*)d_in[4];
    const float* conv_b   = (const float*)d_in[5];
    const float* xproj_w  = (const float*)d_in[6];
    const float* dtproj_w = (const float*)d_in[7];
    const float* dtproj_b = (const float*)d_in[8];
    const float* A_log    = (const float*)d_in[9];
    const float* Dp       = (const float*)d_in[10];
    const float* out_w    = (const float*)d_in[11];
    (void)in_sizes; (void)n_in; (void)out_size; (void)ws_size;

    float* ws = (float*)d_ws;
    size_t off = 0;
    auto alloc = [&](size_t n) { float* p = ws + off; off += (n + 63) & ~(size_t)63; return p; };

    float* hsA   = alloc((size_t)NROWS * CMODEL);      // hs (ping)
    float* hsB   = alloc((size_t)NROWS * CMODEL);      // hs (pong)
    float* resid = alloc((size_t)NROWS * CMODEL);
    float* hn    = alloc((size_t)NROWS * CMODEL);      // reused as dbl after in_proj
    float* xz    = alloc((size_t)NROWS * 2 * DINNER);  // [x | z]; x-half reused for y
    float* xact  = alloc((size_t)NROWS * DINNER);
    float* dtb   = alloc((size_t)NROWS * DINNER);
    float* dbl   = hn;                                  // hn dead once xz is built
    float* ybuf  = xz;                                  // y in x-half, stride 2*DINNER

    float* hs     = hsA;
    float* hs_new = hsB;

    const int nEl = NROWS * CMODEL;
    mad_init_hs<<<(nEl + 255) / 256, 256, 0, stream>>>(x, hs);

    for (int i = 0; i < 2; ++i) {
        if (i == 0)
            mad_axpby<<<(nEl + 255) / 256, 256, 0, stream>>>(resid, hs, hs, 1.0f, 0.0f, nEl);
        else
            mad_axpby<<<(nEl + 255) / 256, 256, 0, stream>>>(resid, resid, hs, 2.0f, 1.0f, nEl);

        for (int d = 0; d < 2; ++d) {
            int bi = 2 * i + d;

            // hn = LN(resid, w[bi], b[bi])
            mad_layernorm<<<NROWS, CMODEL, 0, stream>>>(
                resid, norm_w + bi * CMODEL, norm_b + bi * CMODEL, hn);

            // xz = hn @ in_w[bi]^T   (M=4096, N=1024, K=256)  [16x64 strip/wave]
            mad_wmma_gemm<4><<<dim3((2 * DINNER) / 64, NROWS / 16), 32, 0, stream>>>(
                hn, CMODEL, in_w + (size_t)bi * 2 * DINNER * CMODEL, CMODEL,
                nullptr, xz, 2 * DINNER, CMODEL, 0, 0);

            // xact = silu(causal_conv(xz[:, :512], dir))
            mad_conv_silu<<<(NROWS * DINNER + 255) / 256, 256, 0, stream>>>(
                xz, conv_w + (size_t)bi * DINNER * DCONV, conv_b + bi * DINNER, xact, d);

            // dbl = xact @ xproj_w[bi]^T   (N=48, K=512)  [one 16x48 strip/wave]
            mad_wmma_gemm<3><<<dim3(1, NROWS / 16), 32, 0, stream>>>(
                xact, DINNER, xproj_w + (size_t)bi * DBLW * DINNER, DINNER,
                nullptr, dbl, DBLW, DINNER, 0, 0);

            // dt = softplus(dbl[:, :16] @ dtproj_w[bi]^T + dtproj_b[bi])  (N=512, K=16)
            mad_wmma_gemm<4><<<dim3(DINNER / 64, NROWS / 16), 32, 0, stream>>>(
                dbl, DBLW, dtproj_w + (size_t)bi * DINNER * DTRANK, DTRANK,
                dtproj_b + bi * DINNER, dtb, DINNER, DTRANK, 1, 0);

            // y (into x-half of xz) = selective_scan(...) * silu(z)
            mad_scan<<<(NSEQ * DINNER + 255) / 256, 256, 0, stream>>>(
                xact, dtb, dbl, xz,
                A_log + (size_t)bi * DINNER * DSTATE, Dp + bi * DINNER,
                ybuf, 2 * DINNER, d);

            // hs_new (+)= y @ out_w[bi]^T   (N=256, K=512); accumulate on 2nd dir
            mad_wmma_gemm<4><<<dim3(CMODEL / 64, NROWS / 16), 32, 0, stream>>>(
                ybuf, 2 * DINNER, out_w + (size_t)bi * CMODEL * DINNER, DINNER,
                nullptr, hs_new, CMODEL, DINNER, 0, d);
        }

        float* tmp = hs; hs = hs_new; hs_new = tmp;
    }

    // out = (2*resid + hs) transposed back to (bn, c, l)
    mad_final<<<(nEl + 255) / 256, 256, 0, stream>>>(resid, hs, (float*)d_out);
}